// GNN_Agent_19945828123200
// MI455X (gfx1250) — compile-verified
//
#include <hip/hip_runtime.h>
#include <hip/hip_bf16.h>

// ---------------------------------------------------------------------------
// GNN message passing + GRU update, 3 layers, fp32, on gfx1250 (MI455X).
//   m   = x @ W[l]                       (WMMA f32 16x16x4 GEMM)
//   agg = scatter_add(m[src] -> dst)     (global_atomic_add_f32, L2-resident)
//   x   = GRUCell(agg, x)                (fused 6-accumulator WMMA GEMM + act)
// N=50000 (=16*3125), C=96 (=16*6), E=800000 -> no tile remainders anywhere.
// ---------------------------------------------------------------------------

#define N_NODES 50000
#define N_EDGES 800000
#define CDIM    96
#define NC      (N_NODES * CDIM)            // 4,800,000 floats per node tensor
#define TILES   ((N_NODES / 16) * (CDIM / 16))  // 18750 16x16 output tiles

typedef __attribute__((ext_vector_type(2))) float v2f;
typedef __attribute__((ext_vector_type(8))) float v8f;

// ---------------------------------------------------------------------------
// Kernel 1: m = x @ W   ([N,96] @ [96,96])
// One wave32 computes one 16x16 tile of m; K loop steps by 4 (24 WMMAs).
// A layout (ISA 7.12.2, 32-bit A 16x4): lanes 0-15 hold K=k0,k0+1 ; lanes
// 16-31 hold K=k0+2,k0+3, row M = lane&15.  B/C/D: row striped across lanes.
// ---------------------------------------------------------------------------
__global__ __launch_bounds__(256) void gnn_gemm_m(const float* __restrict__ x,
                                                  const float* __restrict__ W,
                                                  float* __restrict__ m) {
  const int wave = blockIdx.x * (blockDim.x >> 5) + (threadIdx.x >> 5);
  if (wave >= TILES) return;                 // wave-uniform: EXEC stays full
  const int tr   = wave / (CDIM / 16);
  const int tc   = wave % (CDIM / 16);
  const int lane = threadIdx.x & 31;
  const int half = lane >> 4;                // 0: K pair (k0,k0+1), 1: (k0+2,k0+3)
  const int l16  = lane & 15;
  const int row0 = tr * 16, col0 = tc * 16;

  const float* arow = x + (row0 + l16) * CDIM + 2 * half;  // A: row l16
  const float* bcol = W + col0 + l16;                      // B: col l16

  v8f acc = {0.f, 0.f, 0.f, 0.f, 0.f, 0.f, 0.f, 0.f};
#pragma unroll
  for (int k0 = 0; k0 < CDIM; k0 += 4) {
    v2f a, b;
    a.x = arow[k0];
    a.y = arow[k0 + 1];
    const int k = k0 + 2 * half;
    b.x = bcol[(k    ) * CDIM];
    b.y = bcol[(k + 1) * CDIM];
    acc = __builtin_amdgcn_wmma_f32_16x16x4_f32(false, a, false, b,
                                                (short)0, acc, false, false);
  }

  float* orow = m + (row0 + 8 * half) * CDIM + col0 + l16;
#pragma unroll
  for (int r = 0; r < 8; ++r) orow[r * CDIM] = acc[r];
}

// ---------------------------------------------------------------------------
// Kernel 2: zero agg (poisoned workspace -> must clear before atomics)
// ---------------------------------------------------------------------------
__global__ void gnn_zero(float4* __restrict__ p, int n4) {
  int i = blockIdx.x * blockDim.x + threadIdx.x;
  if (i < n4) p[i] = make_float4(0.f, 0.f, 0.f, 0.f);
}

// ---------------------------------------------------------------------------
// Kernel 3: agg[dst] += m[src]  (segment_sum over 800K edges, 96 channels)
// One work item = (edge, 4-channel group): float4 gather (hits 192MB L2) +
// 4 fp32 atomic adds (L2 RMW units).  Grid-stride over 19.2M items.
// ---------------------------------------------------------------------------
__global__ void gnn_scatter(const float* __restrict__ m,
                            const int* __restrict__ edge_index,
                            float* __restrict__ agg) {
  const long long total  = (long long)N_EDGES * (CDIM / 4);
  const long long stride = (long long)gridDim.x * blockDim.x;
  for (long long i = (long long)blockIdx.x * blockDim.x + threadIdx.x;
       i < total; i += stride) {
    const int e = (int)(i / (CDIM / 4));
    const int g = (int)(i % (CDIM / 4));
    const int s = edge_index[e];             // src row  (edge_index[0])
    const int d = edge_index[N_EDGES + e];   // dst row  (edge_index[1])
    const float4 v = *(const float4*)(m + s * CDIM + g * 4);
    float* p = agg + d * CDIM + g * 4;
    atomicAdd(p + 0, v.x);
    atomicAdd(p + 1, v.y);
    atomicAdd(p + 2, v.z);
    atomicAdd(p + 3, v.w);
  }
}

// ---------------------------------------------------------------------------
// Kernel 4: fused GRU cell.  One wave per 16x16 tile of x_new.
// Six WMMA accumulators: gi_{r,z,n} = agg @ w_ih.T (+b_ih),
//                        gh_{r,z,n} = h   @ w_hh.T (+b_hh),
// B[k][n] = w_ih[seg*96 + col][k] (row-major [288,96] -> transpose is just
// a stride-1 read along k).  Bias folded into the accumulator init.
// ---------------------------------------------------------------------------
__global__ __launch_bounds__(256) void gnn_gru(const float* __restrict__ agg,
                                               const float* __restrict__ h,
                                               const float* __restrict__ w_ih,
                                               const float* __restrict__ w_hh,
                                               const float* __restrict__ b_ih,
                                               const float* __restrict__ b_hh,
                                               float* __restrict__ out) {
  const int wave = blockIdx.x * (blockDim.x >> 5) + (threadIdx.x >> 5);
  if (wave >= TILES) return;
  const int tr   = wave / (CDIM / 16);
  const int tc   = wave % (CDIM / 16);
  const int lane = threadIdx.x & 31;
  const int half = lane >> 4;
  const int l16  = lane & 15;
  const int row0 = tr * 16, col0 = tc * 16;
  const int col  = col0 + l16;

  v8f a_ir, a_iz, a_in, a_hr, a_hz, a_hn;
  {
    const float bir = b_ih[col], biz = b_ih[CDIM + col], bin_ = b_ih[2 * CDIM + col];
    const float bhr = b_hh[col], bhz = b_hh[CDIM + col], bhn  = b_hh[2 * CDIM + col];
#pragma unroll
    for (int r = 0; r < 8; ++r) {
      a_ir[r] = bir; a_iz[r] = biz; a_in[r] = bin_;
      a_hr[r] = bhr; a_hz[r] = bhz; a_hn[r] = bhn;
    }
  }

  const float* aggrow = agg + (row0 + l16) * CDIM + 2 * half;
  const float* hrow   = h   + (row0 + l16) * CDIM + 2 * half;
  const float* wih_r  = w_ih + (0 * CDIM + col) * CDIM;
  const float* wih_z  = w_ih + (1 * CDIM + col) * CDIM;
  const float* wih_n  = w_ih + (2 * CDIM + col) * CDIM;
  const float* whh_r  = w_hh + (0 * CDIM + col) * CDIM;
  const float* whh_z  = w_hh + (1 * CDIM + col) * CDIM;
  const float* whh_n  = w_hh + (2 * CDIM + col) * CDIM;

#pragma unroll 4
  for (int k0 = 0; k0 < CDIM; k0 += 4) {
    v2f ai, ah, b;
    ai.x = aggrow[k0]; ai.y = aggrow[k0 + 1];
    ah.x = hrow[k0];   ah.y = hrow[k0 + 1];
    const int k = k0 + 2 * half;

    b.x = wih_r[k]; b.y = wih_r[k + 1];
    a_ir = __builtin_amdgcn_wmma_f32_16x16x4_f32(false, ai, false, b, (short)0, a_ir, false, false);
    b.x = wih_z[k]; b.y = wih_z[k + 1];
    a_iz = __builtin_amdgcn_wmma_f32_16x16x4_f32(false, ai, false, b, (short)0, a_iz, false, false);
    b.x = wih_n[k]; b.y = wih_n[k + 1];
    a_in = __builtin_amdgcn_wmma_f32_16x16x4_f32(false, ai, false, b, (short)0, a_in, false, false);
    b.x = whh_r[k]; b.y = whh_r[k + 1];
    a_hr = __builtin_amdgcn_wmma_f32_16x16x4_f32(false, ah, false, b, (short)0, a_hr, false, false);
    b.x = whh_z[k]; b.y = whh_z[k + 1];
    a_hz = __builtin_amdgcn_wmma_f32_16x16x4_f32(false, ah, false, b, (short)0, a_hz, false, false);
    b.x = whh_n[k]; b.y = whh_n[k + 1];
    a_hn = __builtin_amdgcn_wmma_f32_16x16x4_f32(false, ah, false, b, (short)0, a_hn, false, false);
  }

  // Epilogue: r = sig(i_r+h_r), z = sig(i_z+h_z), n = tanh(i_n + r*h_n),
  //           out = (1-z)*n + z*h
#pragma unroll
  for (int r = 0; r < 8; ++r) {
    const int row = row0 + r + 8 * half;
    const float hv = h[row * CDIM + col];
    const float rr = 1.f / (1.f + __expf(-(a_ir[r] + a_hr[r])));
    const float zz = 1.f / (1.f + __expf(-(a_iz[r] + a_hz[r])));
    const float nn = tanhf(a_in[r] + rr * a_hn[r]);
    out[row * CDIM + col] = (1.f - zz) * nn + zz * hv;
  }
}

// ---------------------------------------------------------------------------
// Launcher: 3 layers ping-ponged through workspace.
// ws layout: xa | xb | m | agg   (4 * 4.8M floats = 76.8 MB)
// ---------------------------------------------------------------------------
extern "C" void kernel_launch(void* const* d_in, const int* in_sizes, int n_in,
                              void* d_out, int out_size, void* d_ws, size_t ws_size,
                              hipStream_t stream) {
  const float* x      = (const float*)d_in[0];
  const int*   ei     = (const int*)  d_in[1];
  const float* weight = (const float*)d_in[2];   // [3,96,96]
  const float* w_ih   = (const float*)d_in[3];   // [288,96]
  const float* w_hh   = (const float*)d_in[4];   // [288,96]
  const float* b_ih   = (const float*)d_in[5];   // [288]
  const float* b_hh   = (const float*)d_in[6];   // [288]
  float* out = (float*)d_out;

  float* xa  = (float*)d_ws;
  float* xb  = xa + NC;
  float* m   = xb + NC;
  float* agg = m  + NC;

  const int wavesPerBlock = 256 / 32;                         // 8
  const int gemmBlocks    = (TILES + wavesPerBlock - 1) / wavesPerBlock;  // 2344
  const int zeroBlocks    = (NC / 4 + 255) / 256;             // 4688
  const int scatBlocks    = 2048;                             // grid-stride

  for (int l = 0; l < 3; ++l) {
    const float* xin  = (l == 0) ? x  : ((l == 1) ? xa : xb);
    float*       xout = (l == 2) ? out : ((l == 0) ? xa : xb);
    const float* Wl   = weight + l * CDIM * CDIM;

    gnn_gemm_m<<<gemmBlocks, 256, 0, stream>>>(xin, Wl, m);
    gnn_zero  <<<zeroBlocks, 256, 0, stream>>>((float4*)agg, NC / 4);
    gnn_scatter<<<scatBlocks, 256, 0, stream>>>(m, ei, agg);
    gnn_gru   <<<gemmBlocks, 256, 0, stream>>>(agg, xin, w_ih, w_hh, b_ih, b_hh, xout);
  }
}